// ViterbiDecoder_429496730259
// MI455X (gfx1250) — compile-verified
//
#include <hip/hip_runtime.h>

// Smoothed Viterbi alignment DP (anti-diagonal wavefront), MI455X / gfx1250.
//
// Critical-path bound: 4095 strictly dependent diagonals. One 1024-thread
// workgroup (32 wave32 waves on a single WGP), three LDS diagonal buffers
// (98 KB of the WGP's 320 KB), one split-barrier per diagonal. Each cell is
// stored as (max, exp(V0-max), exp(V1-max), exp(V2-max)) so each incoming
// logsumexp is one 3-dot + v_log_f32; emitting a cell costs 3x v_exp_f32.
// The own-row dependency (V[i, j-1]) is carried in a register; theta/phi for
// the next diagonal are register-pipelined; deep global_prefetch_b8 warms
// HBM-cold theta lines ~12 diagonals ahead. Loop is unrolled by 3 (4095 % 3
// == 0) so the buffer rotation uses static LDS offsets.

#define VN 2048
#define VMm 2048
#define NDIAG (VN + VMm)       // last diagonal d = N + M = 4096
#define ROWS (VN + 1)          // skew indices 0..2048 (index 0 = boundary row)

__device__ __forceinline__ size_t theta_idx(int i, int d) {
    // theta[i-1, j-1] with j = d - i; clamp so loads are unconditional
    // (invalid cells never consume the value).
    int jm1 = d - i - 1;
    jm1 = jm1 < 0 ? 0 : (jm1 > (VMm - 1) ? (VMm - 1) : jm1);
    return (size_t)(i - 1) * (size_t)VMm + (size_t)jm1;
}

__device__ __forceinline__ int col_idx(int i, int d) {
    int jm1 = d - i - 1;
    return jm1 < 0 ? 0 : (jm1 > (VMm - 1) ? (VMm - 1) : jm1);
}

__device__ __forceinline__ void dp_cell(
    int i, int j, float th, float ps, float ph,
    const float* __restrict__ eA,
    const float4* __restrict__ p2,   // diagonal d-2
    const float4* __restrict__ p1,   // diagonal d-1
    float4* __restrict__ pc,         // diagonal d (output)
    float4& own)                     // this row's value at d-1 (register)
{
    if (j >= 1 && j <= VMm) {
        float4 a2 = p2[i - 1];   // V[i-1, j-1]  (neighbor row, via LDS)
        float4 a1 = p1[i - 1];   // V[i-1, j  ]  (neighbor row, via LDS)
        float4 b1 = own;         // V[i,   j-1]  (own row, register-carried)

        float sm = fmaf(eA[0], a2.y, fmaf(eA[1], a2.z, eA[2] * a2.w));
        float sx = fmaf(eA[3], a1.y, fmaf(eA[4], a1.z, eA[5] * a1.w));
        float sy = fmaf(eA[6], b1.y, fmaf(eA[7], b1.z, eA[8] * b1.w));

        float Vm = th + a2.x + __logf(sm);   // match
        float Vx = ps + a1.x + __logf(sx);   // gap in X
        float Vy = ph + b1.x + __logf(sy);   // gap in Y

        float mx = fmaxf(Vm, fmaxf(Vx, Vy));
        float4 o;
        o.x = mx;
        o.y = __expf(Vm - mx);
        o.z = __expf(Vx - mx);
        o.w = __expf(Vy - mx);
        own = o;
        pc[i] = o;
    } else if (j == 0) {
        // Newly-boundary cell (i == d): V = (0,0,0) -> (mx=0, e=1,1,1).
        // Cells with j < 0 or j > M are provably never read again; `own`
        // already holds the boundary value from initialization.
        pc[i] = make_float4(0.f, 1.f, 1.f, 1.f);
    }
}

__device__ __forceinline__ void diag_step(
    int d,
    const float* __restrict__ theta, const float* __restrict__ phi,
    int i0, int i1, float ps0, float ps1,
    const float* __restrict__ eA,
    const float4* __restrict__ p2, const float4* __restrict__ p1,
    float4* __restrict__ pc,
    float4& c0, float4& c1,
    float& th0, float& th1, float& ph0, float& ph1)
{
    // Next-diagonal operands: issued first, consumed after the barrier.
    float th0n = theta[theta_idx(i0, d + 1)];
    float th1n = theta[theta_idx(i1, d + 1)];
    float ph0n = phi[col_idx(i0, d + 1)];
    float ph1n = phi[col_idx(i1, d + 1)];
    // Deep cache warm-up for HBM-cold theta lines (global_prefetch_b8).
    __builtin_prefetch(theta + theta_idx(i0, d + 12), 0, 0);
    __builtin_prefetch(theta + theta_idx(i1, d + 12), 0, 0);

    dp_cell(i0, d - i0, th0, ps0, ph0, eA, p2, p1, pc, c0);
    dp_cell(i1, d - i1, th1, ps1, ph1, eA, p2, p1, pc, c1);

    __syncthreads();

    th0 = th0n; th1 = th1n; ph0 = ph0n; ph1 = ph1n;
}

__global__ __launch_bounds__(1024, 1) void viterbi_wavefront(
    const float* __restrict__ theta,
    const float* __restrict__ psi,
    const float* __restrict__ phi,
    const float* __restrict__ A,
    float* __restrict__ out)
{
    __shared__ float4 bufA[ROWS];
    __shared__ float4 bufB[ROWS];
    __shared__ float4 bufC[ROWS];

    const int t = threadIdx.x;

    const float4 bnd = make_float4(0.f, 1.f, 1.f, 1.f);
    for (int k = t; k < ROWS; k += 1024) {
        bufA[k] = bnd;
        bufB[k] = bnd;
        bufC[k] = bnd;
    }

    // exp of the 3x3 transition matrix, once per thread (rows are
    // log-softmaxed, so logsumexp(A[m] + 0) == 0 falls out for free).
    float eA[9];
#pragma unroll
    for (int k = 0; k < 9; ++k) eA[k] = __expf(A[k]);

    const int i0 = t + 1;            // skew rows owned by this thread
    const int i1 = t + 1 + 1024;
    const float ps0 = psi[i0 - 1];
    const float ps1 = psi[i1 - 1];

    // Register-carried own-row state (V at d-1 for rows i0 / i1),
    // initialized to the zero boundary.
    float4 c0 = bnd;
    float4 c1 = bnd;

    // Register pipeline prologue: operands for d = 2.
    float th0 = theta[theta_idx(i0, 2)];
    float th1 = theta[theta_idx(i1, 2)];
    float ph0 = phi[col_idx(i0, 2)];
    float ph1 = phi[col_idx(i1, 2)];

    __syncthreads();

    // 4095 diagonals = 1365 groups of 3; static buffer rotation:
    //   d : p2=A p1=B pc=C | d+1 : p2=B p1=C pc=A | d+2 : p2=C p1=A pc=B
#pragma unroll 1
    for (int d = 2; d + 2 <= NDIAG; d += 3) {
        diag_step(d,     theta, phi, i0, i1, ps0, ps1, eA,
                  bufA, bufB, bufC, c0, c1, th0, th1, ph0, ph1);
        diag_step(d + 1, theta, phi, i0, i1, ps0, ps1, eA,
                  bufB, bufC, bufA, c0, c1, th0, th1, ph0, ph1);
        diag_step(d + 2, theta, phi, i0, i1, ps0, ps1, eA,
                  bufC, bufA, bufB, c0, c1, th0, th1, ph0, ph1);
    }

    // Thread 1023 owns row i1 = 2048; after d = 4096 its register state is
    // V[N, M]. Terminal score Vt = logsumexp over the 3 states.
    if (t == 1023) {
        out[0] = c1.x + __logf(c1.y + c1.z + c1.w);
    }
}

extern "C" void kernel_launch(void* const* d_in, const int* in_sizes, int n_in,
                              void* d_out, int out_size, void* d_ws, size_t ws_size,
                              hipStream_t stream) {
    (void)in_sizes; (void)n_in; (void)d_ws; (void)ws_size; (void)out_size;
    const float* theta = (const float*)d_in[0];
    const float* psi   = (const float*)d_in[1];
    const float* phi   = (const float*)d_in[2];
    const float* A     = (const float*)d_in[3];
    float* out = (float*)d_out;

    // Single workgroup: the DP's 4095-step dependent chain must live on one
    // WGP so each step syncs via the in-WGP split barrier + LDS.
    viterbi_wavefront<<<1, 1024, 0, stream>>>(theta, psi, phi, A, out);
}